// BEVFeatureExtractor_25134148616992
// MI455X (gfx1250) — compile-verified
//
#include <hip/hip_runtime.h>
#include <cstdint>

typedef float v4f __attribute__((ext_vector_type(4)));

#define H_DIM 180
#define W_DIM 180
#define C_DIM 512
#define THREADS 128           // C_DIM / 4 channels per thread (float4)
#define PTS_PER_BLOCK 4       // depth-4 async pipeline, 4 LDS buffers

template<int CNT>
__device__ __forceinline__ void wait_async() {
    asm volatile("s_wait_asynccnt %0" :: "i"(CNT) : "memory");
}

__device__ __forceinline__ void async_load16(uint32_t lds_off, const float* gptr) {
    // CDNA5 async copy global -> LDS, tracked by ASYNCcnt.
    asm volatile("global_load_async_to_lds_b128 %0, %1, off"
                 :: "v"(lds_off), "v"((uint64_t)(uintptr_t)gptr)
                 : "memory");
}

// Compute bilinear weights and issue the 4 corner-row gathers for point p.
// Weights are returned; corner offsets are consumed immediately by the issue.
__device__ __forceinline__ void setup_and_issue(const float* __restrict__ feat,
                                                const float* __restrict__ centers,
                                                int p, int b,
                                                uint32_t lds_base, int buf, int tid,
                                                float& wa, float& wb, float& wc, float& wd) {
    float cx = centers[(long long)p * 2 + 0];
    float cy = centers[(long long)p * 2 + 1];
    // mirror reference float math: ((c - PC_START)/VOXEL)/OUT_STRIDE
    float x = ((cx + 54.0f) / 0.075f) / 8.0f;
    float y = ((cy + 54.0f) / 0.075f) / 8.0f;
    int x0 = (int)floorf(x), y0 = (int)floorf(y);
    int x1 = x0 + 1, y1 = y0 + 1;
    x0 = min(max(x0, 0), W_DIM - 1); x1 = min(max(x1, 0), W_DIM - 1);
    y0 = min(max(y0, 0), H_DIM - 1); y1 = min(max(y1, 0), H_DIM - 1);
    float x0f = (float)x0, x1f = (float)x1, y0f = (float)y0, y1f = (float)y1;
    wa = (x1f - x) * (y1f - y);
    wb = (x1f - x) * (y - y0f);
    wc = (x - x0f) * (y1f - y);
    wd = (x - x0f) * (y - y0f);

    long long base = (long long)b * (H_DIM * W_DIM * C_DIM);
    long long oa = base + (long long)(y0 * W_DIM + x0) * C_DIM;
    long long ob = base + (long long)(y1 * W_DIM + x0) * C_DIM;
    long long oc = base + (long long)(y0 * W_DIM + x1) * C_DIM;
    long long od = base + (long long)(y1 * W_DIM + x1) * C_DIM;

    // Each corner row is 512 floats = 2 KB; 128 threads x 16 B cover it exactly.
    uint32_t l = lds_base + (uint32_t)buf * (4u * C_DIM * 4u) + (uint32_t)tid * 16u;
    int c4 = tid * 4;
    async_load16(l + 0u,    feat + oa + c4);
    async_load16(l + 2048u, feat + ob + c4);
    async_load16(l + 4096u, feat + oc + c4);
    async_load16(l + 6144u, feat + od + c4);
}

__global__ __launch_bounds__(THREADS)
void bev_extract_kernel(const float* __restrict__ feat,
                        const float* __restrict__ centers,
                        const int* __restrict__ num_point,
                        float* __restrict__ out,
                        int B, int N) {
    __shared__ __align__(16) float smem[PTS_PER_BLOCK][4][C_DIM]; // 32 KB, 4 buffers
    const int tid   = threadIdx.x;
    const int total = B * N;
    const int np    = *num_point;
    const int sec   = (np > 1) ? (N / np) : N;
    const int p0    = blockIdx.x * PTS_PER_BLOCK;
    if (p0 >= total) return;
    const int nvalid = min(PTS_PER_BLOCK, total - p0);
    const uint32_t lds_base = (uint32_t)(uintptr_t)&smem[0][0][0];

    // Index bookkeeping done once; updated incrementally below (no per-point sdiv).
    int b = p0 / N;
    int n = p0 - b * N;
    int q = (np > 1) ? (n / sec) : 0;
    int s = n - q * sec;

    // ---- Issue phase: put all gathers for this block's points in flight ----
    float wa[PTS_PER_BLOCK], wb[PTS_PER_BLOCK], wc[PTS_PER_BLOCK], wd[PTS_PER_BLOCK];
    {
        int bi = b, ni = n;
        #pragma unroll
        for (int i = 0; i < PTS_PER_BLOCK; ++i) {
            if (i < nvalid) {
                setup_and_issue(feat, centers, p0 + i, bi, lds_base, i, tid,
                                wa[i], wb[i], wc[i], wd[i]);
            }
            if (++ni == N) { ni = 0; ++bi; }
        }
    }

    // ---- Consume phase: staged ASYNCcnt waits (async loads complete in order) ----
    const int c4 = tid * 4;
    #pragma unroll
    for (int i = 0; i < PTS_PER_BLOCK; ++i) {
        if (i < nvalid) {
            if (nvalid == PTS_PER_BLOCK) {
                // fast path: groups i+1..3 may remain in flight
                switch (i) {
                    case 0: wait_async<12>(); break;
                    case 1: wait_async<8>();  break;
                    case 2: wait_async<4>();  break;
                    default: wait_async<0>(); break;
                }
            } else {
                wait_async<0>();
            }

            v4f A  = *(const v4f*)&smem[i][0][c4];
            v4f Bv = *(const v4f*)&smem[i][1][c4];
            v4f Cv = *(const v4f*)&smem[i][2][c4];
            v4f D  = *(const v4f*)&smem[i][3][c4];
            v4f r  = A * wa[i] + Bv * wb[i] + Cv * wc[i] + D * wd[i];

            if (np > 1) {
                if (q < np) { // n < sec*np  <=>  q < np
                    long long dst = (((long long)b * sec + s) * np + q) * C_DIM + c4;
                    __builtin_nontemporal_store(r, (v4f*)(out + dst)); // NT b128 store
                }
            } else {
                long long dst = ((long long)b * N + n) * C_DIM + c4;
                __builtin_nontemporal_store(r, (v4f*)(out + dst));
            }
        }
        // incremental (b, n, q, s) update — replaces per-point divisions
        ++n; ++s;
        if (s == sec) { s = 0; ++q; }
        if (n == N)   { n = 0; ++b; q = 0; s = 0; }
    }
}

extern "C" void kernel_launch(void* const* d_in, const int* in_sizes, int n_in,
                              void* d_out, int out_size, void* d_ws, size_t ws_size,
                              hipStream_t stream) {
    const float* feat    = (const float*)d_in[0];
    const float* centers = (const float*)d_in[1];
    const int* num_point = (const int*)d_in[2];
    float* out = (float*)d_out;

    const int B = in_sizes[0] / (H_DIM * W_DIM * C_DIM);
    const int N = in_sizes[1] / (B * 2);
    const int total  = B * N;
    const int blocks = (total + PTS_PER_BLOCK - 1) / PTS_PER_BLOCK;

    bev_extract_kernel<<<blocks, THREADS, 0, stream>>>(feat, centers, num_point, out, B, N);
}